// Toroidal_KDE_11347303596038
// MI455X (gfx1250) — compile-verified
//
#include <hip/hip_runtime.h>
#include <math.h>

// Toroidal von-Mises-product KDE, reformulated as a rank-6 GEMM + exp2 + row-mean.
//   dot2[t,n] = log2(e) * sum_d k_d cos(c_td - o_nd)        (log2e folded into A)
//   out[t]    = scale * sum_n exp2(dot2[t,n]),  scale = exp(-sum_d log(2*pi*I0(k_d))) / N
//
// Staged fragment rows use K-slot order [0,1,4,5 | 2,3,6,7] so each half-wave's
// four per-lane fragment elements are one aligned float4:
//   lanes 0-15  read slots 0..3  = K{0,1} (frag0) + K{4,5} (frag1)
//   lanes 16-31 read slots 4..7  = K{2,3} (frag0) + K{6,7} (frag1)
// matching the ISA 7.12.2 f32 A(16x4)/B(4x16) VGPR layouts for
// V_WMMA_F32_16X16X4_F32 chained twice (K = 8, real K = 6, rest zero).
//
// ws layout (floats): Afrag[Tpad][8] | Bfrag[Npad][8] | scale[1]
//   needs (Tpad + Npad)*32 + 4 bytes  (~451 KB for T=4096, N=10000)

typedef float v2f __attribute__((ext_vector_type(2)));
typedef float v4f __attribute__((ext_vector_type(4)));
typedef float v8f __attribute__((ext_vector_type(8)));

#define KDE_NWAVES 8
#define KDE_TPB    (KDE_NWAVES * 32)
#define LOG2E_F    1.44269504088896340736f

// ---------------------------------------------------------------- prep ------
__global__ void kde_prep(const float* __restrict__ coords,  // [T][3]
                         const float* __restrict__ kappa,   // [3]
                         const float* __restrict__ obs,     // [N][3]
                         float* __restrict__ Afrag,         // [Tpad][8] permuted K
                         float* __restrict__ Bfrag,         // [Npad][8] permuted K
                         float* __restrict__ scale,         // [1]
                         int T, int Tpad, int N, int Npad) {
  // K-slot permutation: slot s holds K = kmap[s]
  const int kmap[8] = {0, 1, 4, 5, 2, 3, 6, 7};
  int i = blockIdx.x * blockDim.x + threadIdx.x;

  if (i < Tpad) {
    float vals[8] = {0, 0, 0, 0, 0, 0, 0, 0};
    if (i < T) {
#pragma unroll
      for (int d = 0; d < 3; ++d) {
        float kd = kappa[d] * LOG2E_F;     // fold log2(e) into the A features
        float c  = coords[i * 3 + d];
        vals[d]     = kd * cosf(c);
        vals[3 + d] = kd * sinf(c);
      }
    }
    float* a = Afrag + (size_t)i * 8;
#pragma unroll
    for (int s = 0; s < 8; ++s) a[s] = vals[kmap[s]];
  } else if (i < Tpad + Npad) {
    int n = i - Tpad;
    float vals[8] = {0, 0, 0, 0, 0, 0, 0, 0};
    if (n < N) {
#pragma unroll
      for (int d = 0; d < 3; ++d) {
        float o     = obs[n * 3 + d];
        vals[d]     = cosf(o);
        vals[3 + d] = sinf(o);
      }
    }
    float* b = Bfrag + (size_t)n * 8;
#pragma unroll
    for (int s = 0; s < 8; ++s) b[s] = vals[kmap[s]];
  }

  if (i == 0) {
    // scale = exp(-sum_d log(2*pi*I0(kappa_d))) / N   (NORM_TERM == 1)
    double s = 0.0;
#pragma unroll
    for (int d = 0; d < 3; ++d) {
      double x = (double)kappa[d];
      double q = 0.25 * x * x;
      double term = 1.0, i0 = 1.0;
      for (int k = 1; k < 40; ++k) {        // I0 power series, ample for kappa <= 4.5
        term *= q / ((double)k * (double)k);
        i0 += term;
      }
      s += log(6.283185307179586 * i0);
    }
    scale[0] = (float)(exp(-s) / (double)N);
  }
}

// ---------------------------------------------------------------- main ------
// One block per 16-row tile of T. 8 waves stride over the N/16 column tiles.
__global__ void __launch_bounds__(KDE_TPB)
kde_main(const float* __restrict__ Afrag,   // [Tpad][8] permuted K
         const float* __restrict__ Bfrag,   // [Npad][8] permuted K
         const float* __restrict__ scale,   // [1]
         float* __restrict__ out,           // [T]
         int T, int Ntiles, int padcols) {
  __shared__ float rowacc[16];
  if (threadIdx.x < 16) rowacc[threadIdx.x] = 0.0f;
  __syncthreads();

  const int t0   = blockIdx.x * 16;
  const int lane = (int)(threadIdx.x & 31);
  const int wave = (int)(threadIdx.x >> 5);
  const int half = lane >> 4;     // 0: lanes 0-15, 1: lanes 16-31
  const int l15  = lane & 15;

  // A fragment: one aligned float4 per lane (rows are 32B apart, half*16B offset)
  const v4f av = *(const v4f*)(Afrag + (size_t)(t0 + l15) * 8 + half * 4);
  v2f a0, a1;
  a0[0] = av[0]; a0[1] = av[1];
  a1[0] = av[2]; a1[1] = av[3];

  float acc[8];
#pragma unroll
  for (int v = 0; v < 8; ++v) acc[v] = 0.0f;

#pragma unroll 2
  for (int nt = wave; nt < Ntiles; nt += KDE_NWAVES) {
    // B fragment: one aligned float4 per lane
    const v4f bv = *(const v4f*)(Bfrag + (size_t)(nt * 16 + l15) * 8 + half * 4);
    v2f b0, b1;
    b0[0] = bv[0]; b0[1] = bv[1];
    b1[0] = bv[2]; b1[1] = bv[3];

    v8f c = {};
    c = __builtin_amdgcn_wmma_f32_16x16x4_f32(false, a0, false, b0, (short)0, c,
                                              false, false);
    c = __builtin_amdgcn_wmma_f32_16x16x4_f32(false, a1, false, b1, (short)0, c,
                                              false, false);
    // C/D layout: VGPR v -> row (v + 8*half), col = l15 of this tile
#pragma unroll
    for (int v = 0; v < 8; ++v) acc[v] += __builtin_amdgcn_exp2f(c[v]);
  }

  // Reduce across the 16 columns (lanes within each half, xor masks < 16)
#pragma unroll
  for (int m = 1; m < 16; m <<= 1) {
#pragma unroll
    for (int v = 0; v < 8; ++v) acc[v] += __shfl_xor(acc[v], m, 32);
  }

  // One lane per half folds its 8 row-partials into LDS (across the 8 waves)
  if (l15 == 0) {
#pragma unroll
    for (int v = 0; v < 8; ++v) atomicAdd(&rowacc[half * 8 + v], acc[v]);
  }
  __syncthreads();

  if (threadIdx.x < 16) {
    int t = t0 + (int)threadIdx.x;
    if (t < T)
      out[t] = (rowacc[threadIdx.x] - (float)padcols) * scale[0];
  }
}

// ---------------------------------------------------------------- launch ----
extern "C" void kernel_launch(void* const* d_in, const int* in_sizes, int n_in,
                              void* d_out, int out_size, void* d_ws, size_t ws_size,
                              hipStream_t stream) {
  const float* coords = (const float*)d_in[0];   // [T][3]
  const float* kappa  = (const float*)d_in[1];   // [3]
  const float* obs    = (const float*)d_in[2];   // [N][3]
  float*       out    = (float*)d_out;           // [T]

  const int T    = in_sizes[0] / 3;
  const int N    = in_sizes[2] / 3;
  const int Tpad = (T + 15) & ~15;
  const int Npad = (N + 15) & ~15;

  float* ws    = (float*)d_ws;
  float* Afrag = ws;
  float* Bfrag = ws + (size_t)Tpad * 8;
  float* scale = Bfrag + (size_t)Npad * 8;

  const int prepThreads = Tpad + Npad;
  kde_prep<<<(prepThreads + 255) / 256, 256, 0, stream>>>(
      coords, kappa, obs, Afrag, Bfrag, scale, T, Tpad, N, Npad);

  const int Ntiles  = Npad / 16;
  const int padcols = Npad - N;
  kde_main<<<Tpad / 16, KDE_TPB, 0, stream>>>(
      Afrag, Bfrag, scale, out, T, Ntiles, padcols);
}